// Clahe_67070209294628
// MI455X (gfx1250) — compile-verified
//
#include <hip/hip_runtime.h>
#include <hip/hip_bf16.h>
#include <math.h>

// CLAHE: 16 x 1024 x 1024 fp32 image, 8x8 blocks of 128x128, 256 levels, clip 10.0
// Phase 1: per-block 256-bin histogram, float4 streaming loads + LDS atomics
// Phase 2: clip + cumsum + LUT; cumsum as (16x32)x(32x16) matmul via
//          8 chained v_wmma_f32_16x16x4_f32 per wave (1 wave / block), bit-exact
// Phase 3: 8-row x 1024-col slabs; 16 LUT tables (16KB) staged into LDS,
//          per-pixel bilinear blend gathers from LDS (VMEM stays streaming)

#define NB    16          // batch
#define HDIM  1024
#define WDIM  1024
#define BLK   8           // blocks per side
#define BM    128         // block height
#define BN    128         // block width
#define LVL   256
#define TV    640.0f      // (BM*BN)/LVL * 10.0
#define NBLOCKS (NB*BLK*BLK)   // 1024

typedef __attribute__((ext_vector_type(2))) float v2f;
typedef __attribute__((ext_vector_type(8))) float v8f;

// ---------------------------------------------------------------- histogram
__global__ void clahe_hist_kernel(const float* __restrict__ img,
                                  float* __restrict__ hist) {
    __shared__ int h[8][LVL];                 // per-wave privatized histograms
    const int tid  = threadIdx.x;             // 0..255
    const int wave = tid >> 5;                // 8 waves of 32

    #pragma unroll
    for (int i = tid; i < 8 * LVL; i += 256) ((int*)h)[i] = 0;
    __syncthreads();

    const int blk = blockIdx.x;               // b*64 + by*8 + bx
    const int b  = blk >> 6;
    const int by = (blk >> 3) & 7;
    const int bx = blk & 7;
    const float* base = img + ((size_t)b << 20) + (size_t)by * BM * WDIM + (size_t)bx * BN;

    // 16384 px / (256 thr * 4 px) = 16 iters of float4, coalesced along rows
    #pragma unroll 2
    for (int k = 0; k < 16; ++k) {
        int lin  = (k << 8) + tid;            // float4 unit id, 0..4095
        int row  = lin >> 5;                  // 32 float4 per 128-px row
        int col4 = lin & 31;
        float4 p = *(const float4*)(base + (size_t)row * WDIM + (col4 << 2));
        atomicAdd(&h[wave][(int)p.x], 1);     // ds_add_u32
        atomicAdd(&h[wave][(int)p.y], 1);
        atomicAdd(&h[wave][(int)p.z], 1);
        atomicAdd(&h[wave][(int)p.w], 1);
    }
    __syncthreads();

    int s = 0;
    #pragma unroll
    for (int w = 0; w < 8; ++w) s += h[w][tid];
    hist[(size_t)blk * LVL + tid] = (float)s;
}

// ------------------------------------------------- clip + WMMA cumsum + LUT
__global__ void clahe_cdf_kernel(const float* __restrict__ hist,
                                 float* __restrict__ maps) {
    __shared__ float sh_clip[LVL];
    __shared__ float sh_g[16];

    const int blk  = blockIdx.x;              // one wave (32 lanes) per block
    const int lane = threadIdx.x;             // 0..31
    const int half = lane >> 4;               // 0 or 1
    const int st   = lane & 15;               // s for A rows / t for B cols

    const float* hb = hist + (size_t)blk * LVL;

    // each lane owns 8 consecutive bins: compute clip excess
    float hv[8];
    float extra = 0.0f;
    #pragma unroll
    for (int i = 0; i < 8; ++i) {
        hv[i] = hb[lane * 8 + i];
        extra += fmaxf(hv[i] - TV, 0.0f);
    }
    // wave32 reduction for total_extra
    #pragma unroll
    for (int m = 16; m > 0; m >>= 1) extra += __shfl_xor(extra, m, 32);
    const float me = extra * (1.0f / 256.0f);

    #pragma unroll
    for (int i = 0; i < 8; ++i) {
        float c = (hv[i] >= TV) ? (TV + me) : (hv[i] + me);
        sh_clip[lane * 8 + i] = floorf(c);
    }
    __syncthreads();

    // per-segment row sums g[u] (all lanes compute redundantly -> no divergence)
    {
        float g = 0.0f;
        #pragma unroll
        for (int v = 0; v < 16; ++v) g += sh_clip[st * 16 + v];
        sh_g[st] = g;                         // lanes L and L+16 write same value
    }
    __syncthreads();

    // cdf = A' x B', A' = [Lstrict | Clip] (16x32), B' = [G ; Utri] (32x16)
    // K=32 split into 8 chunks of 4 -> 8 x v_wmma_f32_16x16x4_f32
    v8f acc = {};
    #pragma unroll
    for (int q = 0; q < 8; ++q) {
        const int k0 = 4 * q + 2 * half;      // this lane-half's K pair
        float a0, a1, b0, b1;
        if (q < 4) {                          // Lstrict columns / G rows
            a0 = (k0     < st) ? 1.0f : 0.0f;
            a1 = (k0 + 1 < st) ? 1.0f : 0.0f;
            b0 = sh_g[k0];
            b1 = sh_g[k0 + 1];
        } else {                              // Clip columns / upper-tri rows
            const int kk = k0 - 16;
            a0 = sh_clip[st * 16 + kk];
            a1 = sh_clip[st * 16 + kk + 1];
            b0 = (kk     <= st) ? 1.0f : 0.0f;
            b1 = (kk + 1 <= st) ? 1.0f : 0.0f;
        }
        v2f a = {a0, a1};
        v2f b = {b0, b1};
        acc = __builtin_amdgcn_wmma_f32_16x16x4_f32(
            /*neg_a=*/false, a, /*neg_b=*/false, b,
            /*c_mod=*/(short)0, acc, /*reuse_a=*/false, /*reuse_b=*/false);
    }

    // D layout: VGPR r holds cdf[r + 8*half, t] ; maps = floor(cdf*255/16384)%256
    float* mo = maps + (size_t)blk * LVL;
    #pragma unroll
    for (int r = 0; r < 8; ++r) {
        const int s = r + 8 * half;
        int m = ((int)floorf(acc[r] * (255.0f / 16384.0f))) & 255;
        mo[s * 16 + st] = (float)m;
    }
}

// ------------------------------------------------------------- apply (blend)
// One workgroup per 8-row x 1024-col slab (r-band constant: band edges at
// 192, 320, ... are multiples of 64, so aligned 8-row slabs never cross).
__global__ void clahe_apply_kernel(const float* __restrict__ img,
                                   const float* __restrict__ maps,
                                   float* __restrict__ out) {
    __shared__ float sh[16 * LVL];            // 16 tables x 256 = 16 KB
    const int tid   = threadIdx.x;            // 0..255
    const int chunk = blockIdx.x;             // 0..2047
    const int b  = chunk >> 7;                // image
    const int i0 = (chunk & 127) << 3;        // base row of 8-row slab

    // row-band terms (uniform over the slab)
    const int   r   = (int)truncf(((float)i0 - 64.0f) * (1.0f / 128.0f));
    const int   rp  = min(r + 1, BLK - 1);
    const bool  rlast = (r >= BLK - 1);

    // stage 16 LUT tables: t<8 -> (r, t), t>=8 -> (rp, t-8); float4 coalesced
    const float4* mb4 = (const float4*)(maps + ((size_t)b << 14));
    float4* sh4 = (float4*)sh;
    #pragma unroll
    for (int u = tid; u < 1024; u += 256) {
        int t  = u >> 6;                      // table id
        int q  = u & 63;                      // float4 within table
        int rr = (t < 8) ? r : rp;
        int cc = t & 7;
        sh4[u] = mb4[(((rr << 3) + cc) << 6) + q];
    }
    __syncthreads();

    // column terms (fixed per thread: 4 aligned pixels share c)
    const int   j0  = tid << 2;
    const float fj0 = (float)j0;
    const int   c   = (int)truncf((fj0 - 64.0f) * (1.0f / 128.0f));
    const int   cp  = min(c + 1, BLK - 1);
    const bool  clast = (c >= BLK - 1);
    const float* s_lu = sh + (c << 8);
    const float* s_lb = sh + ((8 + c) << 8);
    const float* s_ru = sh + (cp << 8);
    const float* s_rb = sh + ((8 + cp) << 8);

    float y1e[4];
    #pragma unroll
    for (int u = 0; u < 4; ++u) {
        float y1 = ((fj0 + (float)u) - ((float)c + 0.5f) * 128.0f) * (1.0f / 128.0f);
        y1e[u] = clast ? 0.0f : y1;
    }

    const float* src = img + ((size_t)b << 20) + ((size_t)i0 << 10) + j0;
    float*       dst = out + ((size_t)b << 20) + ((size_t)i0 << 10) + j0;

    #pragma unroll
    for (int ri = 0; ri < 8; ++ri) {
        const float fi  = (float)(i0 + ri);
        const float x1  = (fi - ((float)r + 0.5f) * 128.0f) * (1.0f / 128.0f);
        const float x1e = rlast ? 0.0f : x1;

        const float4 pv = *(const float4*)(src + ((size_t)ri << 10));
        const float pvv[4] = {pv.x, pv.y, pv.z, pv.w};
        float res[4];
        #pragma unroll
        for (int u = 0; u < 4; ++u) {
            const int v = (int)pvv[u];
            const float lu = s_lu[v], lb = s_lb[v], ru = s_ru[v], rb = s_rb[v];
            float o = (1.0f - y1e[u]) * ((1.0f - x1e) * lu + x1e * lb)
                    +          y1e[u] * ((1.0f - x1e) * ru + x1e * rb);
            int t = (int)truncf(o);
            t %= 256;
            if (t < 0) t += 256;              // Python-style mod
            res[u] = (float)t;
        }
        *(float4*)(dst + ((size_t)ri << 10)) = make_float4(res[0], res[1], res[2], res[3]);
    }
}

// ---------------------------------------------------------------- launcher
extern "C" void kernel_launch(void* const* d_in, const int* in_sizes, int n_in,
                              void* d_out, int out_size, void* d_ws, size_t ws_size,
                              hipStream_t stream) {
    const float* img = (const float*)d_in[0];
    float* out  = (float*)d_out;
    float* hist = (float*)d_ws;                    // 1024*256 floats = 1 MB
    float* maps = hist + (size_t)NBLOCKS * LVL;    // 1024*256 floats = 1 MB

    hipLaunchKernelGGL(clahe_hist_kernel,  dim3(NBLOCKS), dim3(256), 0, stream, img, hist);
    hipLaunchKernelGGL(clahe_cdf_kernel,   dim3(NBLOCKS), dim3(32),  0, stream, hist, maps);

    // 16 images * 128 slabs (8 rows x 1024 cols each)
    hipLaunchKernelGGL(clahe_apply_kernel, dim3(NB * 128), dim3(256), 0, stream,
                       img, maps, out);
}